// HyperGraphEmbed_73031623901536
// MI455X (gfx1250) — compile-verified
//
#include <hip/hip_runtime.h>
#include <hip/hip_bf16.h>
#include <math.h>

// ---------------------------------------------------------------------------
// HyperGraph GNN forward for MI455X (gfx1250, wave32).
// GEMMs use V_WMMA_F32_16X16X4_F32 (exact fp32 matrix path).
// Scatters use wave-per-edge fp32 global atomics (L2-resident destinations).
// ---------------------------------------------------------------------------

#define EM 128

typedef float v2f __attribute__((ext_vector_type(2)));
typedef float v8f __attribute__((ext_vector_type(8)));

static __device__ __forceinline__ void atomAddF(float* p, float v) {
  unsafeAtomicAdd(p, v);  // lowers to global_atomic_add_f32 on gfx1250
}

// ---------------------------------------------------------------------------
// GEMM: Y[N,128] = X[N,128] @ W[128,128]   (mode 0)
//       Y = sigmoid(X@W + bias) * gateX    (mode 1, gating epilogue)
// One wave -> one 16x16 tile of D. 8 waves/block cover all 128 columns for a
// 16-row strip. K looped 4 at a time through V_WMMA_F32_16X16X4_F32.
//
// fp32 WMMA fragment layouts (CDNA5 ISA 7.12.2):
//  A 16x4 : lanes 0-15 hold {K0,K1} of row M=lane; lanes 16-31 hold {K2,K3}.
//  B 4x16 : lanes 0-15 hold {K0,K1} of col N=lane; lanes 16-31 hold {K2,K3}.
//  C/D    : vgpr m holds row (m + 8*half), col = lane&15.
//
// Out-of-range A rows are CLAMPED to row 0 (always valid) and loaded
// unconditionally -> no EXEC divergence in the k-loop; the garbage only
// feeds output rows that the store-side guard drops.
// ---------------------------------------------------------------------------
__global__ __launch_bounds__(256) void gemm128_kernel(
    const float* __restrict__ X, const float* __restrict__ W,
    const float* __restrict__ bias, const float* __restrict__ gateX,
    float* __restrict__ Y, int nrows, int mode) {
  const int lane = threadIdx.x & 31;
  const int wave = threadIdx.x >> 5;
  const int half = lane >> 4;           // 0 or 1
  const int l15  = lane & 15;
  const int rowBase = blockIdx.x * 16;
  const int colBase = wave * 16;

  const int arow = rowBase + l15;
  const int crow = (arow < nrows) ? arow : 0;          // clamp: always valid
  const float* xrow = X + (size_t)crow * EM + half * 2;
  const float* wcol = W + (size_t)(half * 2) * EM + colBase + l15;

  v8f acc = {};
#pragma unroll
  for (int k = 0; k < EM; k += 4) {
    const v2f a = *(const v2f*)(xrow + k);
    v2f b;
    b.x = wcol[(size_t)k * EM];
    b.y = wcol[(size_t)(k + 1) * EM];
    acc = __builtin_amdgcn_wmma_f32_16x16x4_f32(
        false, a, false, b, (short)0, acc, false, false);
  }

  const int col = colBase + l15;
#pragma unroll
  for (int m = 0; m < 8; ++m) {
    const int row = rowBase + m + half * 8;
    if (row < nrows) {
      float v = acc[m];
      if (mode == 1) {
        v += bias[col];
        const float s = 1.0f / (1.0f + __expf(-v));
        v = s * gateX[(size_t)row * EM + col];
      }
      Y[(size_t)row * EM + col] = v;
    }
  }
}

// ---------------------------------------------------------------------------
// Wave-per-edge row scatter:  dst[didx[e]] += w(e) * src[sidx[e]]
// w(e) = (ev?ev[e]:1) * (rsS?rsS[sidx]:1) * (rsD?rsD[didx]:1)
// 32 lanes x float4 = one 128-float row per wave; fp32 global atomics.
// ---------------------------------------------------------------------------
__global__ __launch_bounds__(256) void scatter_rows_kernel(
    float* __restrict__ dst, const float* __restrict__ src,
    const int* __restrict__ sidx, const int* __restrict__ didx,
    const float* __restrict__ ev, const float* __restrict__ rsS,
    const float* __restrict__ rsD, int nnz) {
  const int e = blockIdx.x * (blockDim.x >> 5) + (threadIdx.x >> 5);
  if (e >= nnz) return;
  const int lane = threadIdx.x & 31;
  const int s = sidx[e];
  const int d = didx[e];
  float w = 1.0f;
  if (ev)  w *= ev[e];
  if (rsS) w *= rsS[s];
  if (rsD) w *= rsD[d];
  const float4 v = ((const float4*)(src + (size_t)s * EM))[lane];
  float* p = dst + (size_t)d * EM + lane * 4;
  atomAddF(p + 0, v.x * w);
  atomAddF(p + 1, v.y * w);
  atomAddF(p + 2, v.z * w);
  atomAddF(p + 3, v.w * w);
}

// deg[idx[i]] += 1
__global__ void count_kernel(float* __restrict__ deg, const int* __restrict__ idx,
                             int nnz) {
  const int i = blockIdx.x * blockDim.x + threadIdx.x;
  if (i < nnz) atomAddF(&deg[idx[i]], 1.0f);
}

// deg[idx[i]] += val[i]
__global__ void val_count_kernel(float* __restrict__ deg,
                                 const int* __restrict__ idx,
                                 const float* __restrict__ val, int nnz) {
  const int i = blockIdx.x * blockDim.x + threadIdx.x;
  if (i < nnz) atomAddF(&deg[idx[i]], val[i]);
}

// mode 0: safe_inv(x)   mode 1: (x+1)^-1/2   (deg with implicit self-loop)
__global__ void finalize_kernel(float* __restrict__ d, int n, int mode) {
  const int i = blockIdx.x * blockDim.x + threadIdx.x;
  if (i >= n) return;
  const float x = d[i];
  d[i] = (mode == 0) ? (x > 0.0f ? 1.0f / x : 0.0f) : rsqrtf(x + 1.0f);
}

// dst[r][:] = src[r][:] * rs[r]   (sq: * rs[r]^2)
__global__ void scale_rows_kernel(float* __restrict__ dst,
                                  const float* __restrict__ src,
                                  const float* __restrict__ rs, int rows, int sq) {
  const int i = blockIdx.x * blockDim.x + threadIdx.x;
  if (i >= rows * EM) return;
  float s = rs[i >> 7];
  if (sq) s *= s;
  dst[i] = src[i] * s;
}

// x = leaky_relu(Y*(rs?rs[row]:1) + bias[col]) + x ;  res += coef*x
__global__ void combine_kernel(const float* __restrict__ Y,
                               const float* __restrict__ rs,
                               const float* __restrict__ bias,
                               float* __restrict__ x, float* __restrict__ res,
                               float coef, int rows) {
  const int i = blockIdx.x * blockDim.x + threadIdx.x;
  if (i >= rows * EM) return;
  float v = Y[i];
  if (rs) v *= rs[i >> 7];
  v += bias[i & 127];
  v = v > 0.0f ? v : 0.01f * v;              // leaky_relu(0.01)
  const float xn = v + x[i];
  x[i] = xn;
  res[i] += coef * xn;
}

// ---------------------------------------------------------------------------
static inline int cdiv(int a, int b) { return (a + b - 1) / b; }

extern "C" void kernel_launch(void* const* d_in, const int* in_sizes, int n_in,
                              void* d_out, int out_size, void* d_ws, size_t ws_size,
                              hipStream_t stream) {
  (void)in_sizes; (void)n_in; (void)out_size; (void)ws_size;

  const int ND = 20000, NP = 15000, NM = 5000;
  const int NI = 40000;   // nodes
  const int NE = 30000;   // hyperedges / A nodes
  const int NH = 600000, NA = 600000, NDD = 100000;

  const float* disE = (const float*)d_in[0];
  const float* proE = (const float*)d_in[1];
  const float* medE = (const float*)d_in[2];
  const int*   Hx   = (const int*)d_in[3];    // [0:NH]=nidx, [NH:2NH]=eidx
  const int*   Ax   = (const int*)d_in[4];    // [0:NA]=row,  [NA:2NA]=col
  const float* Aval = (const float*)d_in[5];
  const int*   Dx   = (const int*)d_in[6];    // [0:NDD]=row, [NDD:2NDD]=col
  const float* hgW  = (const float*)d_in[7];
  const float* hgB  = (const float*)d_in[8];
  const float* lgW  = (const float*)d_in[9];
  const float* lgB  = (const float*)d_in[10];
  const float* hW   = (const float*)d_in[11]; // [2,128,128]
  const float* hB   = (const float*)d_in[12];
  const float* gW   = (const float*)d_in[13];
  const float* gB   = (const float*)d_in[14];
  const float* dW   = (const float*)d_in[15];
  const float* dB   = (const float*)d_in[16];

  float* w = (float*)d_ws;
  size_t off = 0;
  float* W0 = w + off; off += (size_t)NI * EM;  // embed / ddi-Y / hgat-Y
  float* W1 = w + off; off += (size_t)NI * EM;  // ddi-x / h_x
  float* W2 = w + off; off += (size_t)NI * EM;  // l_embed / xw (all GEMMs)
  float* W4 = w + off; off += (size_t)NE * EM;  // ef / gat-Y
  float* W5 = w + off; off += (size_t)NE * EM;  // l_rep
  float* W6 = w + off; off += (size_t)NE * EM;  // g_x
  float* sDinvN = w + off; off += NI;           // 1/node-degree (H)
  float* sBinvE = w + off; off += NE;           // 1/edge-degree (H)
  float* sDisA  = w + off; off += NE;           // deg(A)^-1/2
  float* sDisD  = w + off; off += NM;           // deg(ddi)^-1/2

  float* out      = (float*)d_out;
  float* outHyper = out;                                   // [NE,128]
  float* outLin   = out + (size_t)NE * EM;                 // [NE,128]
  float* outH     = out + (size_t)2 * NE * EM;             // [NI,128]
  float* outDdi   = out + (size_t)2 * NE * EM + (size_t)NI * EM; // [NM,128]

  const size_t rowB = (size_t)EM * sizeof(float);

  // ================= ddi GCN stack (independent, runs first) ================
  hipMemsetAsync(sDisD, 0, NM * sizeof(float), stream);
  count_kernel<<<cdiv(NDD, 256), 256, 0, stream>>>(sDisD, Dx + NDD, NDD);
  finalize_kernel<<<cdiv(NM, 256), 256, 0, stream>>>(sDisD, NM, 1);
  hipMemcpyAsync(W1, medE, (size_t)NM * rowB, hipMemcpyDeviceToDevice, stream);
  hipMemcpyAsync(outDdi, medE, (size_t)NM * rowB, hipMemcpyDeviceToDevice, stream);
  for (int i = 0; i < 2; ++i) {
    gemm128_kernel<<<cdiv(NM, 16), 256, 0, stream>>>(
        W1, dW + (size_t)i * EM * EM, nullptr, nullptr, W2, NM, 0);
    scale_rows_kernel<<<cdiv(NM * EM, 256), 256, 0, stream>>>(W0, W2, sDisD, NM, 1);
    scatter_rows_kernel<<<cdiv(NDD, 8), 256, 0, stream>>>(
        W0, W2, Dx, Dx + NDD, nullptr, sDisD, sDisD, NDD);
    combine_kernel<<<cdiv(NM * EM, 256), 256, 0, stream>>>(
        W0, nullptr, dB + (size_t)i * EM, W1, outDdi, 1.0f / (i + 2), NM);
  }

  // ================= build embed = concat(dis, pro, med) ====================
  hipMemcpyAsync(W0, disE, (size_t)ND * rowB, hipMemcpyDeviceToDevice, stream);
  hipMemcpyAsync(W0 + (size_t)ND * EM, proE, (size_t)NP * rowB,
                 hipMemcpyDeviceToDevice, stream);
  hipMemcpyAsync(W0 + (size_t)(ND + NP) * EM, medE, (size_t)NM * rowB,
                 hipMemcpyDeviceToDevice, stream);

  // ================= degrees ================================================
  hipMemsetAsync(sDinvN, 0, NI * sizeof(float), stream);
  count_kernel<<<cdiv(NH, 256), 256, 0, stream>>>(sDinvN, Hx, NH);
  finalize_kernel<<<cdiv(NI, 256), 256, 0, stream>>>(sDinvN, NI, 0);

  hipMemsetAsync(sBinvE, 0, NE * sizeof(float), stream);
  count_kernel<<<cdiv(NH, 256), 256, 0, stream>>>(sBinvE, Hx + NH, NH);
  finalize_kernel<<<cdiv(NE, 256), 256, 0, stream>>>(sBinvE, NE, 0);

  hipMemsetAsync(sDisA, 0, NE * sizeof(float), stream);
  val_count_kernel<<<cdiv(NA, 256), 256, 0, stream>>>(sDisA, Ax + NA, Aval, NA);
  finalize_kernel<<<cdiv(NE, 256), 256, 0, stream>>>(sDisA, NE, 1);

  // ================= gates (WMMA GEMM + fused sigmoid gate) =================
  gemm128_kernel<<<cdiv(NI, 16), 256, 0, stream>>>(W0, hgW, hgB, W0, W1, NI, 1);
  gemm128_kernel<<<cdiv(NI, 16), 256, 0, stream>>>(W0, lgW, lgB, W0, W2, NI, 1);

  // res_h starts as gated h_embed (lands directly in d_out)
  hipMemcpyAsync(outH, W1, (size_t)NI * rowB, hipMemcpyDeviceToDevice, stream);

  // ================= l_rep = mean_{v in e} l_embed[v] =======================
  hipMemsetAsync(W5, 0, (size_t)NE * rowB, stream);
  scatter_rows_kernel<<<cdiv(NH, 8), 256, 0, stream>>>(
      W5, W2, Hx, Hx + NH, nullptr, nullptr, nullptr, NH);
  scale_rows_kernel<<<cdiv(NE * EM, 256), 256, 0, stream>>>(W5, W5, sBinvE, NE, 0);
  hipMemcpyAsync(W6, W5, (size_t)NE * rowB, hipMemcpyDeviceToDevice, stream);
  hipMemcpyAsync(outLin, W5, (size_t)NE * rowB, hipMemcpyDeviceToDevice, stream);

  // ================= hypergraph stack: D^-1 H B^-1 H^T (xW) + b =============
  for (int i = 0; i < 2; ++i) {
    gemm128_kernel<<<cdiv(NI, 16), 256, 0, stream>>>(
        W1, hW + (size_t)i * EM * EM, nullptr, nullptr, W2, NI, 0);
    hipMemsetAsync(W4, 0, (size_t)NE * rowB, stream);
    scatter_rows_kernel<<<cdiv(NH, 8), 256, 0, stream>>>(          // H^T
        W4, W2, Hx, Hx + NH, nullptr, nullptr, nullptr, NH);
    hipMemsetAsync(W0, 0, (size_t)NI * rowB, stream);
    scatter_rows_kernel<<<cdiv(NH, 8), 256, 0, stream>>>(          // H (B^-1 folded)
        W0, W4, Hx + NH, Hx, nullptr, sBinvE, nullptr, NH);
    combine_kernel<<<cdiv(NI * EM, 256), 256, 0, stream>>>(
        W0, sDinvN, hB + (size_t)i * EM, W1, outH, 1.0f / (i + 2), NI);
  }

  // ================= hyper_rep = mean_{v in e} h_embed[v] ===================
  hipMemsetAsync(outHyper, 0, (size_t)NE * rowB, stream);
  scatter_rows_kernel<<<cdiv(NH, 8), 256, 0, stream>>>(
      outHyper, outH, Hx, Hx + NH, nullptr, nullptr, nullptr, NH);
  scale_rows_kernel<<<cdiv(NE * EM, 256), 256, 0, stream>>>(
      outHyper, outHyper, sBinvE, NE, 0);

  // ================= GCN stack on A (l_rep graph) ===========================
  for (int i = 0; i < 2; ++i) {
    gemm128_kernel<<<cdiv(NE, 16), 256, 0, stream>>>(
        W6, gW + (size_t)i * EM * EM, nullptr, nullptr, W2, NE, 0);
    scale_rows_kernel<<<cdiv(NE * EM, 256), 256, 0, stream>>>(     // self loop: dis^2
        W4, W2, sDisA, NE, 1);
    scatter_rows_kernel<<<cdiv(NA, 8), 256, 0, stream>>>(
        W4, W2, Ax, Ax + NA, Aval, sDisA, sDisA, NA);
    combine_kernel<<<cdiv(NE * EM, 256), 256, 0, stream>>>(
        W4, nullptr, gB + (size_t)i * EM, W6, outLin, 1.0f / (i + 2), NE);
  }
}